// WastewaterGAT_47124381172460
// MI455X (gfx1250) — compile-verified
//
#include <hip/hip_runtime.h>
#include <hip/hip_bf16.h>

typedef float v2f __attribute__((ext_vector_type(2)));
typedef float v8f __attribute__((ext_vector_type(8)));

#define HEADS 4
#define HID 16
#define HC 64

// ---------- helpers: order-preserving float<->uint key for atomic max ----------
__device__ __forceinline__ unsigned fkey(float f) {
    unsigned u = __float_as_uint(f);
    return (u & 0x80000000u) ? ~u : (u | 0x80000000u);
}
__device__ __forceinline__ float funkey(unsigned u) {
    if (u == 0u) return 0.0f;  // sentinel: no edge touched this node -> m := 0 (matches isfinite guard)
    return (u & 0x80000000u) ? __uint_as_float(u & 0x7FFFFFFFu) : __uint_as_float(~u);
}

// ---------- generic zero ----------
__global__ void zero_k(float* p, long n) {
    long i = (long)blockIdx.x * blockDim.x + threadIdx.x;
    if (i < n) p[i] = 0.0f;
}

// ---------- precompute folded attention vectors ----------
// attB[K,16]: cols 0-3 = W·att_s per head, cols 4-7 = W·att_d per head, cols 8-15 = 0
// Ve[8,4]:    Ve[k,h] = sum_c We[k, h*16+c] * att_e[h,c]
__global__ void precompute_k(const float* W1, const float* as1, const float* ad1,
                             const float* We1, const float* ae1,
                             const float* W2, const float* as2, const float* ad2,
                             const float* We2, const float* ae2,
                             float* attB1, float* attB2, float* Ve1, float* Ve2) {
    int tid = threadIdx.x;
    for (int i = tid; i < 32 * 16; i += blockDim.x) {
        int k = i >> 4, col = i & 15;
        float v = 0.0f;
        if (col < 8) {
            int h = col & 3;
            const float* att = (col < 4) ? as1 : ad1;
            for (int c = 0; c < HID; ++c) v += W1[k * HC + h * HID + c] * att[h * HID + c];
        }
        attB1[i] = v;
    }
    for (int i = tid; i < 64 * 16; i += blockDim.x) {
        int k = i >> 4, col = i & 15;
        float v = 0.0f;
        if (col < 8) {
            int h = col & 3;
            const float* att = (col < 4) ? as2 : ad2;
            for (int c = 0; c < HID; ++c) v += W2[k * HC + h * HID + c] * att[h * HID + c];
        }
        attB2[i] = v;
    }
    for (int i = tid; i < 8 * 4; i += blockDim.x) {
        int k = i >> 2, h = i & 3;
        float v = 0.0f;
        for (int c = 0; c < HID; ++c) v += We1[k * HC + h * HID + c] * ae1[h * HID + c];
        Ve1[i] = v;
    }
    for (int i = tid; i < 8 * 4; i += blockDim.x) {
        int k = i >> 2, h = i & 3;
        float v = 0.0f;
        for (int c = 0; c < HID; ++c) v += We2[k * HC + h * HID + c] * ae2[h * HID + c];
        Ve2[i] = v;
    }
}

// ---------- node projection GEMM via V_WMMA_F32_16X16X4_F32 ----------
// in[N,K] @ W[K,64] -> h[N,64]; 5th WMMA N-tile with attB gives a_src[N,4], a_dst[N,4].
// One wave = one 16-row tile. W/attB staged through LDS in K-PAIR-INTERLEAVED layout:
//   sW[((k>>1)*64 + col)*2 + (k&1)]  so each lane's (K, K+1) B-values are one aligned
//   8-byte ds_load_b64 directly into the even VGPR pair the WMMA consumes (no shuffles).
__global__ __launch_bounds__(256) void gemm_node_k(
    const float* __restrict__ in, const float* __restrict__ W,
    const float* __restrict__ attB, float* __restrict__ h,
    float* __restrict__ asrc, float* __restrict__ adst, int N, int K) {
    __shared__ float sW[64 * 64];   // [K/2][64][2]
    __shared__ float sB[64 * 16];   // [K/2][16][2]
    int tid = threadIdx.x;
    for (int i = tid; i < K * 64; i += 256) {
        int k = i >> 6, col = i & 63;
        sW[(((k >> 1) << 6) + col) * 2 + (k & 1)] = W[i];
    }
    for (int i = tid; i < K * 16; i += 256) {
        int k = i >> 4, col = i & 15;
        sB[(((k >> 1) << 4) + col) * 2 + (k & 1)] = attB[i];
    }
    __syncthreads();

    int wave = tid >> 5, lane = tid & 31;
    int l16 = lane & 15, hi = lane >> 4;
    long rowBase = ((long)blockIdx.x * 8 + wave) * 16;
    if (rowBase >= N) return;                 // wave-uniform exit: EXEC stays all-1s for WMMA
    long arow = rowBase + l16;
    if (arow >= N) arow = N - 1;              // clamp loads (duplicate reads harmless)

    v8f acc0 = {}, acc1 = {}, acc2 = {}, acc3 = {}, acc4 = {};
    for (int kk = 0; kk < K; kk += 4) {
        int ko = kk + 2 * hi;                 // f32 A/B: lanes 0-15 hold K=ko,ko+1; lanes 16-31 hold ko+2,ko+3
        int p = ko >> 1;                      // K-pair index (ko is even)
        v2f a = *(const v2f*)(&in[arow * K + ko]);
        v2f b;
        b = *(const v2f*)(&sW[((p << 6) + 0 + l16) * 2]);
        acc0 = __builtin_amdgcn_wmma_f32_16x16x4_f32(false, a, false, b, (short)0, acc0, false, false);
        b = *(const v2f*)(&sW[((p << 6) + 16 + l16) * 2]);
        acc1 = __builtin_amdgcn_wmma_f32_16x16x4_f32(false, a, false, b, (short)0, acc1, false, false);
        b = *(const v2f*)(&sW[((p << 6) + 32 + l16) * 2]);
        acc2 = __builtin_amdgcn_wmma_f32_16x16x4_f32(false, a, false, b, (short)0, acc2, false, false);
        b = *(const v2f*)(&sW[((p << 6) + 48 + l16) * 2]);
        acc3 = __builtin_amdgcn_wmma_f32_16x16x4_f32(false, a, false, b, (short)0, acc3, false, false);
        b = *(const v2f*)(&sB[((p << 4) + l16) * 2]);
        acc4 = __builtin_amdgcn_wmma_f32_16x16x4_f32(false, a, false, b, (short)0, acc4, false, false);
    }

    // C/D layout: VGPR v, lanes 0-15 -> row v, lanes 16-31 -> row v+8; col = lane%16 (+16*tile)
#pragma unroll
    for (int v = 0; v < 8; ++v) {
        long r = rowBase + v + 8 * hi;
        if (r < N) {
            h[r * 64 + 0  + l16] = acc0[v];
            h[r * 64 + 16 + l16] = acc1[v];
            h[r * 64 + 32 + l16] = acc2[v];
            h[r * 64 + 48 + l16] = acc3[v];
            float av = acc4[v];
            if (l16 < 4)      asrc[r * 4 + l16]     = av;
            else if (l16 < 8) adst[r * 4 + l16 - 4] = av;
        }
    }
}

// ---------- edge pass 1: alpha = leaky_relu(a_src[s]+a_dst[d]+a_e), atomic segment-max ----------
__global__ void alpha_k(const float* __restrict__ ea, const float* __restrict__ Ve,
                        const float* __restrict__ asrc, const float* __restrict__ adst,
                        const int* __restrict__ src, const int* __restrict__ dst,
                        float* __restrict__ alpha, unsigned* __restrict__ mkey, long E) {
    long e = (long)blockIdx.x * blockDim.x + threadIdx.x;
    if (e >= E) return;
    long s = src[e], d = dst[e];
    float ae0 = 0, ae1 = 0, ae2 = 0, ae3 = 0;
#pragma unroll
    for (int k = 0; k < 8; ++k) {
        float v = ea[e * 8 + k];
        ae0 += v * Ve[k * 4 + 0]; ae1 += v * Ve[k * 4 + 1];
        ae2 += v * Ve[k * 4 + 2]; ae3 += v * Ve[k * 4 + 3];
    }
    float a[4];
    a[0] = asrc[s * 4 + 0] + adst[d * 4 + 0] + ae0;
    a[1] = asrc[s * 4 + 1] + adst[d * 4 + 1] + ae1;
    a[2] = asrc[s * 4 + 2] + adst[d * 4 + 2] + ae2;
    a[3] = asrc[s * 4 + 3] + adst[d * 4 + 3] + ae3;
#pragma unroll
    for (int hh = 0; hh < 4; ++hh) {
        float al = a[hh] > 0.0f ? a[hh] : 0.2f * a[hh];
        alpha[e * 4 + hh] = al;
        atomicMax(&mkey[d * 4 + hh], fkey(al));
    }
}

// ---------- edge pass 2: ex = exp(alpha - m[dst]); den[dst] += ex ----------
__global__ void exp_k(float* __restrict__ alpha, const unsigned* __restrict__ mkey,
                      const int* __restrict__ dst, float* __restrict__ den, long E) {
    long i = (long)blockIdx.x * blockDim.x + threadIdx.x;
    if (i >= E * 4) return;
    long e = i >> 2; int hh = (int)(i & 3);
    long d = dst[e];
    float m = funkey(mkey[d * 4 + hh]);
    float ex = __expf(alpha[i] - m);
    alpha[i] = ex;
    atomicAdd(&den[d * 4 + hh], ex);
}

// ---------- edge pass 3: agg[dst] += (ex/den) * h[src]; float4 gather, 4 atomics ----------
__global__ void agg_k(const float* __restrict__ ex, const float* __restrict__ den,
                      const float* __restrict__ h, const int* __restrict__ src,
                      const int* __restrict__ dst, float* __restrict__ agg, long E) {
    long i = (long)blockIdx.x * blockDim.x + threadIdx.x;
    if (i >= E * 16) return;
    long e = i >> 4; int q = (int)(i & 15); int head = q >> 2;
    long s = src[e], d = dst[e];
    float w = ex[e * 4 + head] / (den[d * 4 + head] + 1e-16f);
    const float4 hv = *(const float4*)(&h[s * 64 + q * 4]);
    float* ap = &agg[d * 64 + q * 4];
    atomicAdd(ap + 0, w * hv.x);
    atomicAdd(ap + 1, w * hv.y);
    atomicAdd(ap + 2, w * hv.z);
    atomicAdd(ap + 3, w * hv.w);
}

// ---------- per-channel sum / sumsq for BatchNorm ----------
__global__ __launch_bounds__(256) void bnstats_k(const float* __restrict__ xin,
                                                 float* __restrict__ stats, int N) {
    __shared__ float ssum[64], ssq[64];
    int tid = threadIdx.x;
    if (tid < 64) { ssum[tid] = 0.0f; ssq[tid] = 0.0f; }
    __syncthreads();
    int c = tid & 63;
    float s = 0.0f, q = 0.0f;
    for (long r = (long)blockIdx.x * 4 + (tid >> 6); r < N; r += (long)gridDim.x * 4) {
        float v = xin[r * 64 + c];
        s += v; q += v * v;
    }
    atomicAdd(&ssum[c], s);
    atomicAdd(&ssq[c], q);
    __syncthreads();
    if (tid < 64) {
        atomicAdd(&stats[tid], ssum[tid]);
        atomicAdd(&stats[64 + tid], ssq[tid]);
    }
}

// ---------- BN apply + ELU (bias b1/b2 cancels inside BN: omitted) ----------
__global__ void bnapply_k(const float* __restrict__ xin, const float* __restrict__ stats,
                          const float* __restrict__ g, const float* __restrict__ b,
                          float* __restrict__ out, long N) {
    long i = (long)blockIdx.x * blockDim.x + threadIdx.x;
    if (i >= N * 64) return;
    int c = (int)(i & 63);
    float invN = 1.0f / (float)N;
    float mu = stats[c] * invN;
    float var = stats[64 + c] * invN - mu * mu;
    float y = (xin[i] - mu) * rsqrtf(var + 1e-5f) * g[c] + b[c];
    out[i] = y > 0.0f ? y : expm1f(y);
}

// ---------- final linear [N,64]@[64,4]+b ----------
__global__ void final_k(const float* __restrict__ xin, const float* __restrict__ Wl,
                        const float* __restrict__ bl, float* __restrict__ out, int N) {
    long n = (long)blockIdx.x * blockDim.x + threadIdx.x;
    if (n >= N) return;
    float o0 = bl[0], o1 = bl[1], o2 = bl[2], o3 = bl[3];
#pragma unroll 8
    for (int k = 0; k < 64; ++k) {
        float v = xin[n * 64 + k];
        o0 += v * Wl[k * 4 + 0]; o1 += v * Wl[k * 4 + 1];
        o2 += v * Wl[k * 4 + 2]; o3 += v * Wl[k * 4 + 3];
    }
    float* op = &out[n * 4];
    op[0] = o0; op[1] = o1; op[2] = o2; op[3] = o3;
}

extern "C" void kernel_launch(void* const* d_in, const int* in_sizes, int n_in,
                              void* d_out, int out_size, void* d_ws, size_t ws_size,
                              hipStream_t stream) {
    const float* x     = (const float*)d_in[0];
    const int*   eidx  = (const int*)d_in[1];
    const float* eattr = (const float*)d_in[2];
    const float* W1    = (const float*)d_in[3];
    const float* as1   = (const float*)d_in[4];
    const float* ad1   = (const float*)d_in[5];
    const float* We1   = (const float*)d_in[6];
    const float* ae1   = (const float*)d_in[7];
    const float* W2    = (const float*)d_in[9];
    const float* as2   = (const float*)d_in[10];
    const float* ad2   = (const float*)d_in[11];
    const float* We2   = (const float*)d_in[12];
    const float* ae2   = (const float*)d_in[13];
    const float* bn1g  = (const float*)d_in[15];
    const float* bn1b  = (const float*)d_in[16];
    const float* bn2g  = (const float*)d_in[17];
    const float* bn2b  = (const float*)d_in[18];
    const float* Wlin  = (const float*)d_in[19];
    const float* blin  = (const float*)d_in[20];

    const int  N = in_sizes[0] / 32;
    const long E = (long)in_sizes[1] / 2;
    const int* src = eidx;
    const int* dst = eidx + E;

    // workspace layout (floats)
    float* ws   = (float*)d_ws;
    float* attB1 = ws;              ws += 32 * 16;
    float* attB2 = ws;              ws += 64 * 16;
    float* Ve1   = ws;              ws += 32;
    float* Ve2   = ws;              ws += 32;
    float* stats = ws;              ws += 128;
    float* hproj = ws;              ws += (long)N * 64;
    float* asrc  = ws;              ws += (long)N * 4;
    float* adst  = ws;              ws += (long)N * 4;
    float* alpha = ws;              ws += E * 4;
    unsigned* mkey = (unsigned*)ws; ws += (long)N * 4;
    float* den   = ws;              ws += (long)N * 4;
    float* agg   = ws;              ws += (long)N * 64;
    float* post  = ws;              ws += (long)N * 64;

    const int B = 256;
    int rowTiles = (N + 15) / 16;
    int gemmBlocks = (rowTiles + 7) / 8;

    precompute_k<<<1, 128, 0, stream>>>(W1, as1, ad1, We1, ae1, W2, as2, ad2, We2, ae2,
                                        attB1, attB2, Ve1, Ve2);

    for (int layer = 0; layer < 2; ++layer) {
        const float* inF  = (layer == 0) ? x : post;
        const float* Wl   = (layer == 0) ? W1 : W2;
        const float* attB = (layer == 0) ? attB1 : attB2;
        const float* Ve   = (layer == 0) ? Ve1 : Ve2;
        const float* bg   = (layer == 0) ? bn1g : bn2g;
        const float* bb   = (layer == 0) ? bn1b : bn2b;
        float* outF       = (layer == 0) ? post : agg;   // layer 2 BN applies in-place
        int K             = (layer == 0) ? 32 : 64;

        zero_k<<<((long)N * 4 + B - 1) / B, B, 0, stream>>>((float*)mkey, (long)N * 4);
        zero_k<<<((long)N * 4 + B - 1) / B, B, 0, stream>>>(den, (long)N * 4);
        zero_k<<<((long)N * 64 + B - 1) / B, B, 0, stream>>>(agg, (long)N * 64);
        zero_k<<<1, 128, 0, stream>>>(stats, 128);

        gemm_node_k<<<gemmBlocks, B, 0, stream>>>(inF, Wl, attB, hproj, asrc, adst, N, K);
        alpha_k<<<(E + B - 1) / B, B, 0, stream>>>(eattr, Ve, asrc, adst, src, dst, alpha, mkey, E);
        exp_k<<<(E * 4 + B - 1) / B, B, 0, stream>>>(alpha, mkey, dst, den, E);
        agg_k<<<(E * 16 + B - 1) / B, B, 0, stream>>>(alpha, den, hproj, src, dst, agg, E);
        bnstats_k<<<1024, B, 0, stream>>>(agg, stats, N);
        bnapply_k<<<((long)N * 64 + B - 1) / B, B, 0, stream>>>(agg, stats, bg, bb, outF, N);
    }

    final_k<<<(N + B - 1) / B, B, 0, stream>>>(agg, Wlin, blin, (float*)d_out, N);
}